// RSNorm_60533269070266
// MI455X (gfx1250) — compile-verified
//
#include <hip/hip_runtime.h>
#include <hip/hip_bf16.h>

typedef __attribute__((ext_vector_type(2))) float v2f;
typedef __attribute__((ext_vector_type(8))) float v8f;

#define DCOLS 256
#define NROWS 131072
#define EPS_V 1e-5f
#define VFLOOR 0.001f

// ws layout (floats): [0,256) colsum  [256,512) colsumsq  [512,768) mu  [768,1024) inv_std

__global__ void zero_ws_kernel(float* __restrict__ ws) {
    int i = blockIdx.x * blockDim.x + threadIdx.x;
    if (i < 2 * DCOLS) ws[i] = 0.0f;
}

// Stage 1: per-column sum and sum-of-squares via V_WMMA_F32_16X16X4_F32.
// A = ones(16x4)  =>  D[m][n] += sum_k B[k][n]  (all 16 output rows identical).
// blockDim = 512 (16 waves); wave w owns columns [16w, 16w+16); block owns a row slab.
__global__ void __launch_bounds__(512) colstat_wmma_kernel(const float* __restrict__ x,
                                                           float* __restrict__ ws) {
    const int lane = threadIdx.x & 31;
    const int wave = threadIdx.x >> 5;          // 0..15 -> column tile
    const int col  = (wave << 4) + (lane & 15); // N = lane & 15 per C/D layout
    const int hi   = lane >> 4;                 // which half of the wave

    const int rows_per_block = NROWS / gridDim.x;   // 256 for grid=512
    const int r0 = blockIdx.x * rows_per_block;

    v2f ones; ones.x = 1.0f; ones.y = 1.0f;
    v8f acc_s = {};   // column sums
    v8f acc_q = {};   // column sums of squares

    // lanes 0-15 supply rows r, r+1 ; lanes 16-31 supply rows r+2, r+3
    const float* base = x + (size_t)(r0 + hi * 2) * DCOLS + col;

#pragma unroll 4
    for (int r = 0; r < rows_per_block; r += 4) {
        const float* p = base + (size_t)r * DCOLS;
        v2f b, bq;
        b.x  = p[0];
        b.y  = p[DCOLS];
        bq.x = b.x * b.x;
        bq.y = b.y * b.y;
        // 8-arg form: (neg_a, A, neg_b, B, c_mod, C, reuse_a, reuse_b)
        acc_s = __builtin_amdgcn_wmma_f32_16x16x4_f32(false, ones, false, b,
                                                      (short)0, acc_s, false, false);
        acc_q = __builtin_amdgcn_wmma_f32_16x16x4_f32(false, ones, false, bq,
                                                      (short)0, acc_q, false, false);
    }

    // Every output row of D is identical -> VGPR0 in every lane holds its column's total.
    float vs = acc_s[0];
    float vq = acc_q[0];
    if (lane < 16) {
        atomicAdd(&ws[col], vs);            // partial column sum
    } else {
        atomicAdd(&ws[DCOLS + col], vq);    // partial column sum of squares
    }
}

// Stage 2: mu, inv_std per column (biased variance with floor, per reference).
__global__ void finalize_stats_kernel(float* __restrict__ ws) {
    const int c = threadIdx.x;
    const float n = (float)NROWS;
    float mu  = ws[c] / n;
    float var = ws[DCOLS + c] / n - mu * mu;
    var = fmaxf(var, VFLOOR);
    ws[2 * DCOLS + c] = mu;
    ws[3 * DCOLS + c] = rsqrtf(var + EPS_V);
}

// Stage 3: streaming normalize, b128 traffic. Stride is a multiple of 256 floats,
// so each lane's 4 columns are fixed -> load mu/inv_std once.
__global__ void __launch_bounds__(256) normalize_kernel(const float* __restrict__ x,
                                                        const float* __restrict__ ws,
                                                        float* __restrict__ out) {
    const size_t tid = (size_t)blockIdx.x * blockDim.x + threadIdx.x;
    const int c4 = (int)((tid * 4) & (DCOLS - 1));
    const float4 mu  = *(const float4*)(ws + 2 * DCOLS + c4);
    const float4 inv = *(const float4*)(ws + 3 * DCOLS + c4);

    const size_t total4 = (size_t)NROWS * DCOLS / 4;
    const size_t stride = (size_t)gridDim.x * blockDim.x;   // 262144 -> cols invariant
    const float4* __restrict__ xin = (const float4*)x;
    float4* __restrict__ o = (float4*)out;

    for (size_t i = tid; i < total4; i += stride) {
        float4 v = xin[i];
        float4 r;
        r.x = (v.x - mu.x) * inv.x;
        r.y = (v.y - mu.y) * inv.y;
        r.z = (v.z - mu.z) * inv.z;
        r.w = (v.w - mu.w) * inv.w;
        o[i] = r;
    }
}

extern "C" void kernel_launch(void* const* d_in, const int* in_sizes, int n_in,
                              void* d_out, int out_size, void* d_ws, size_t ws_size,
                              hipStream_t stream) {
    (void)in_sizes; (void)n_in; (void)out_size; (void)ws_size;
    const float* x = (const float*)d_in[0];
    float* out = (float*)d_out;
    float* ws  = (float*)d_ws;

    zero_ws_kernel<<<2, 256, 0, stream>>>(ws);
    colstat_wmma_kernel<<<512, 512, 0, stream>>>(x, ws);     // 512 blocks x 16 waves
    finalize_stats_kernel<<<1, DCOLS, 0, stream>>>(ws);
    normalize_kernel<<<1024, 256, 0, stream>>>(x, ws, out);
}